// EfficientAttention_68977174774082
// MI455X (gfx1250) — compile-verified
//
#include <hip/hip_runtime.h>

typedef float v2f __attribute__((ext_vector_type(2)));
typedef float v8f __attribute__((ext_vector_type(8)));

#define EA_W 5
#define EA_B 4
#define EA_H 128
#define EA_WW 160
#define EA_C 64
#define EA_M 10
#define EA_K 640          // 5 taps * 128 in-channels
#define EA_EPS 1e-3f

// Repack conv kernel, HWIO flat [k*10+m] -> BpT[m][k] (m padded 10->16 with zeros).
__global__ void ea_prep_kernel(const float* __restrict__ kflat, float* __restrict__ bpt) {
    int idx = blockIdx.x * blockDim.x + threadIdx.x;
    if (idx >= 16 * EA_K) return;
    int m = idx / EA_K;
    int k = idx % EA_K;
    bpt[idx] = (m < EA_M) ? kflat[k * EA_M + m] : 0.0f;
}

__global__ __launch_bounds__(128)
void ea_main_kernel(const float* __restrict__ L, const float* __restrict__ R,
                    const float* __restrict__ bpt, const float* __restrict__ bias,
                    const float* __restrict__ gamma, const float* __restrict__ beta,
                    const float* __restrict__ mean, const float* __restrict__ var,
                    float* __restrict__ out) {
    const int tid  = threadIdx.x;
    const int lane = tid & 31;
    const int wib  = tid >> 5;                    // wave in block (0..3)
    const int tile = blockIdx.x * 4 + wib;        // 16-pixel tile index
    const int lo = lane & 15;
    const int hi = lane >> 4;

    const int p0 = tile * 16;
    const int n  = p0 / (EA_H * EA_WW);
    const int hw = p0 % (EA_H * EA_WW);
    const int h  = hw / EA_WW;
    const int w0 = hw % EA_WW;                    // tile-aligned, 16 | 160 -> same row
    const int w  = w0 + lo;                       // this lane's pixel column
    const size_t rowBase = (size_t)(n * EA_H + h) * EA_WW * EA_C;
    const float* Lrow = L + rowBase;
    const float* Rrow = R + rowBase;

    // LDS: per-wave query tile [20 pos][128 ch], block-shared B [16 m][640 k]
    __shared__ float qtile[4][20][128];           // 40 KB
    __shared__ float bsm[16][640];                // 40 KB

    // ---- stage B: 2560 float4s, 128 threads x 20 ----
    {
        const float4* bsrc = (const float4*)bpt;
        float4* bdst = (float4*)&bsm[0][0];
        #pragma unroll
        for (int it = 0; it < 20; ++it)
            bdst[it * 128 + tid] = bsrc[it * 128 + tid];
    }
    // ---- stage query tile: per wave, 20 positions x 128 ch, zero-padded ----
    {
        float4* qdst = (float4*)&qtile[wib][0][0];
        const int c4 = lane * 4;                  // concat channel 0..124
        #pragma unroll
        for (int it = 0; it < 20; ++it) {
            const int wq = w0 - 2 + it;           // SAME padding window
            float4 v = {0.0f, 0.0f, 0.0f, 0.0f};
            if (wq >= 0 && wq < EA_WW) {
                const float* s = (c4 < EA_C) ? (Lrow + (size_t)wq * EA_C + c4)
                                             : (Rrow + (size_t)wq * EA_C + (c4 - EA_C));
                v = *(const float4*)s;
            }
            qdst[it * 32 + lane] = v;
        }
    }
    __syncthreads();

    // ---- conv as GEMM: D[16 pix][16 m] += A[16 pix][4 k] * B[4 k][16 m] ----
    // A lds addr = (lo*128 + step*4 + hi*2)*4 : linear in step -> imm offsets.
    const v2f* aV = (const v2f*)&qtile[wib][lo][hi * 2];   // +2 v2f per step
    const v2f* bV = (const v2f*)&bsm[lo][hi * 2];          // +2 v2f per step
    v8f acc0 = {}, acc1 = {};
    #pragma unroll
    for (int step = 0; step < EA_K / 4; step += 2) {
        const v2f a0 = aV[2 * step];
        const v2f b0 = bV[2 * step];
        acc0 = __builtin_amdgcn_wmma_f32_16x16x4_f32(false, a0, false, b0,
                                                     (short)0, acc0, false, false);
        const v2f a1 = aV[2 * step + 2];
        const v2f b1 = bV[2 * step + 2];
        acc1 = __builtin_amdgcn_wmma_f32_16x16x4_f32(false, a1, false, b1,
                                                     (short)0, acc1, false, false);
    }
    const v8f acc = acc0 + acc1;

    // ---- D fragment -> LDS so each lane sees its pixel's 10 logits ----
    __shared__ float smem[4][16][16];
    #pragma unroll
    for (int v = 0; v < 8; ++v)
        smem[wib][v + 8 * hi][lo] = acc[v];
    __syncthreads();

    float wv[EA_M];
    float mx = -3.0e38f;
    #pragma unroll
    for (int m = 0; m < EA_M; ++m) {
        wv[m] = smem[wib][lo][m] + bias[m];
        mx = fmaxf(mx, wv[m]);
    }
    float sum = 0.0f;
    #pragma unroll
    for (int m = 0; m < EA_M; ++m) { wv[m] = __expf(wv[m] - mx); sum += wv[m]; }
    const float inv = 1.0f / sum;
    #pragma unroll
    for (int m = 0; m < EA_M; ++m) wv[m] *= inv;

    const int p = p0 + lo;
    const size_t XN = (size_t)EA_B * EA_H * EA_WW * EA_C;
    const size_t WN = (size_t)EA_B * EA_H * EA_WW * EA_M;
    if (hi == 0) {                                // one lane per pixel writes weights
        float* ow  = out + XN + (size_t)p * EA_M;
        float* ow2 = out + XN + WN + (size_t)p * EA_M;
        #pragma unroll
        for (int m = 0; m < EA_M; ++m) { ow[m] = wv[m]; ow2[m] = wv[m]; }
    }

    // ---- softmax-weighted gather over rolled shifts (circular wrap) ----
    const int kb = hi * 32;                       // this lane's 32 channels
    float x[32];
    #pragma unroll
    for (int j = 0; j < 32; ++j) x[j] = 0.0f;
    #pragma unroll
    for (int i = 0; i < EA_W; ++i) {
        int wl = w - i; if (wl < 0)      wl += EA_WW;   // roll(L, +i)
        int wr = w + i; if (wr >= EA_WW) wr -= EA_WW;   // roll(R, -i)
        const float4* ls = (const float4*)(Lrow + (size_t)wl * EA_C + kb);
        const float4* rs = (const float4*)(Rrow + (size_t)wr * EA_C + kb);
        const float wlw = wv[2 * i];
        const float wrw = wv[2 * i + 1];
        #pragma unroll
        for (int q = 0; q < 8; ++q) {
            const float4 lv = ls[q];
            const float4 rv = rs[q];
            x[4 * q + 0] += wlw * lv.x + wrw * rv.x;
            x[4 * q + 1] += wlw * lv.y + wrw * rv.y;
            x[4 * q + 2] += wlw * lv.z + wrw * rv.z;
            x[4 * q + 3] += wlw * lv.w + wrw * rv.w;
        }
    }

    // ---- BatchNorm (inference) + tanh + store ----
    float* ox = out + (size_t)p * EA_C + kb;
    #pragma unroll
    for (int j = 0; j < 32; ++j) {
        const int k = kb + j;
        const float sc = gamma[k] * rsqrtf(var[k] + EA_EPS);
        const float y  = (x[j] - mean[k]) * sc + beta[k];
        ox[j] = tanhf(y);
    }
}

extern "C" void kernel_launch(void* const* d_in, const int* in_sizes, int n_in,
                              void* d_out, int out_size, void* d_ws, size_t ws_size,
                              hipStream_t stream) {
    const float* L     = (const float*)d_in[0];
    const float* R     = (const float*)d_in[1];
    const float* kconv = (const float*)d_in[2];
    const float* bias  = (const float*)d_in[3];
    const float* gamma = (const float*)d_in[4];
    const float* beta  = (const float*)d_in[5];
    const float* mean  = (const float*)d_in[6];
    const float* var   = (const float*)d_in[7];
    float* out = (float*)d_out;
    float* bpt = (float*)d_ws;                    // 16*640*4 = 40960 bytes

    ea_prep_kernel<<<(16 * EA_K + 255) / 256, 256, 0, stream>>>(kconv, bpt);

    const int tiles = EA_B * EA_H * EA_WW / 16;   // 5120 tiles, 4 per block
    ea_main_kernel<<<tiles / 4, 128, 0, stream>>>(L, R, bpt, bias, gamma, beta,
                                                  mean, var, out);
}